// GATLayer_80315888435660
// MI455X (gfx1250) — compile-verified
//
#include <hip/hip_runtime.h>
#include <math.h>

#define BATCH   8
#define NNODES  4096
#define FIN     128
#define NH      4
#define HD      32
#define TOPK    16
#define NEG_SLOPE 0.2f
#define LN_EPS    1e-5f

typedef __attribute__((ext_vector_type(2))) float v2f;
typedef __attribute__((ext_vector_type(8))) float v8f;

// ---------------------------------------------------------------------------
// Kernel 1: h = x @ W  via V_WMMA_F32_16X16X4_F32 (fp32 matrix pipe).
// One wave per 16x16 output tile; K-loop of 32 accumulating WMMAs.
// A 16x4 layout: lane&15 = row M; lanes<16 hold K=0,1 ; lanes>=16 hold K=2,3.
// B 4x16 layout: lane&15 = col N; same half-wave K split.
// C/D: VGPR v -> M = v + 8*(lane>=16), N = lane&15.
// ---------------------------------------------------------------------------
__global__ __launch_bounds__(256) void gemm_xw_wmma(const float* __restrict__ x,
                                                    const float* __restrict__ W,
                                                    float* __restrict__ h) {
  const int lane = threadIdx.x & 31;
  const int wave = threadIdx.x >> 5;
  const int tile = blockIdx.x * 8 + wave;   // 2048*8 = 16384 tiles total
  const int m0 = (tile >> 3) * 16;          // row tile
  const int n0 = (tile & 7) * 16;           // col tile (128/16 = 8)

  const int row = lane & 15;
  const int col = lane & 15;
  const int kb  = (lane >> 4) << 1;         // 0 or 2

  const float* __restrict__ xrow = x + (size_t)(m0 + row) * FIN;

  v8f c = {};
  #pragma unroll 8
  for (int k = 0; k < FIN; k += 4) {
    v2f a, b;
    a.x = xrow[k + kb];
    a.y = xrow[k + kb + 1];
    b.x = W[(size_t)(k + kb)     * FIN + n0 + col];
    b.y = W[(size_t)(k + kb + 1) * FIN + n0 + col];
    c = __builtin_amdgcn_wmma_f32_16x16x4_f32(false, a, false, b,
                                              (short)0, c, false, false);
  }

  const int mbase = m0 + ((lane >> 4) << 3);
  #pragma unroll
  for (int v = 0; v < 8; ++v) {
    h[(size_t)(mbase + v) * FIN + n0 + col] = c[v];
  }
}

// ---------------------------------------------------------------------------
// Kernel 2: per-(node,head) attention logits e_i, e_j (dot over D=32).
// ---------------------------------------------------------------------------
__global__ __launch_bounds__(256) void compute_e(const float* __restrict__ h,
                                                 const float* __restrict__ a_src,
                                                 const float* __restrict__ a_dst,
                                                 float* __restrict__ e_i,
                                                 float* __restrict__ e_j) {
  const int tid  = blockIdx.x * 256 + threadIdx.x;  // BN*NH threads
  const int node = tid >> 2;
  const int head = tid & 3;
  const float* __restrict__ hp = h + (size_t)node * FIN + head * HD;
  const float* __restrict__ as = a_src + head * HD;
  const float* __restrict__ ad = a_dst + head * HD;
  float si = 0.f, sj = 0.f;
  #pragma unroll
  for (int d = 0; d < HD; ++d) {
    float v = hp[d];
    si += v * as[d];
    sj += v * ad[d];
  }
  e_i[tid] = si;
  e_j[tid] = sj;
}

// ---------------------------------------------------------------------------
// Kernel 3: top-16 nearest neighbors per node (brute force over N=4096).
// 256 queries per block (one batch), candidate tiles staged in LDS.
// Strict-< insertion over ascending m matches jax.lax.top_k tie-breaking.
// ---------------------------------------------------------------------------
__global__ __launch_bounds__(256) void topk_kernel(const float* __restrict__ positions,
                                                   int* __restrict__ idx_out) {
  const int b = blockIdx.x >> 4;        // 16 chunks of 256 queries per batch
  const int chunk = blockIdx.x & 15;
  const int n = chunk * 256 + threadIdx.x;

  const float2* __restrict__ pos = (const float2*)positions + (size_t)b * NNODES;
  const float2 q = pos[n];

  float bd[TOPK];
  int   bi[TOPK];
  #pragma unroll
  for (int i = 0; i < TOPK; ++i) { bd[i] = 3.0e38f; bi[i] = 0; }

  __shared__ float2 sp[256];
  const int NT = NNODES / 256;
  for (int t = 0; t < NT; ++t) {
    __syncthreads();
    sp[threadIdx.x] = pos[t * 256 + threadIdx.x];
    if (t + 1 < NT)
      __builtin_prefetch(&pos[(t + 1) * 256 + threadIdx.x], 0, 1);  // global_prefetch_b8
    __syncthreads();

    for (int mm = 0; mm < 256; ++mm) {
      const float dx = q.x - sp[mm].x;
      const float dy = q.y - sp[mm].y;
      const float d2 = dx * dx + dy * dy;
      if (d2 < bd[TOPK - 1]) {
        float v = d2;
        int  vi = t * 256 + mm;
        #pragma unroll
        for (int i = 0; i < TOPK; ++i) {
          if (v < bd[i]) {
            float tf = bd[i]; bd[i] = v;  v  = tf;
            int   ti = bi[i]; bi[i] = vi; vi = ti;
          }
        }
      }
    }
  }

  int* __restrict__ op = idx_out + ((size_t)b * NNODES + n) * TOPK;
  #pragma unroll
  for (int i = 0; i < TOPK; ++i) op[i] = bi[i];
}

// ---------------------------------------------------------------------------
// Kernel 4: gather + leaky-relu scores + softmax(k) + weighted sum +
// residual + LayerNorm. One 128-thread block (4 wave32) per node;
// thread f owns output feature f, head = f/32.
// ---------------------------------------------------------------------------
__global__ __launch_bounds__(128) void aggregate_kernel(
    const float* __restrict__ x,    const float* __restrict__ mask,
    const float* __restrict__ h,    const float* __restrict__ e_i,
    const float* __restrict__ e_j,  const int* __restrict__ idx,
    const float* __restrict__ gamma,const float* __restrict__ beta,
    float* __restrict__ out) {
  const int node = blockIdx.x;          // b*N + n
  const int b    = node / NNODES;
  const int f    = threadIdx.x;         // 0..127
  const int head = f >> 5;

  __shared__ int   s_idx[TOPK];
  __shared__ float s_attn[NH * TOPK];
  __shared__ float s_red[4];
  __shared__ float s_stat;

  if (f < TOPK) s_idx[f] = idx[(size_t)node * TOPK + f];
  __syncthreads();

  // scores (note: reference sums e_i and e_j BOTH at the neighbor index)
  if (f < NH * TOPK) {
    const int kk = f & 15, hh = f >> 4;
    const int gnode = b * NNODES + s_idx[kk];
    float s = e_i[gnode * NH + hh] + e_j[gnode * NH + hh];
    s = (s > 0.f) ? s : NEG_SLOPE * s;            // leaky_relu
    if (mask[gnode] == 0.f) s = -1e9f;
    s_attn[hh * TOPK + kk] = s;
  }
  __syncthreads();

  // softmax over k, one thread per head
  if (f < NH) {
    float mx = -3.0e38f;
    #pragma unroll
    for (int k = 0; k < TOPK; ++k) mx = fmaxf(mx, s_attn[f * TOPK + k]);
    float tmp[TOPK];
    float sum = 0.f;
    #pragma unroll
    for (int k = 0; k < TOPK; ++k) {
      float e = __expf(s_attn[f * TOPK + k] - mx);
      tmp[k] = e; sum += e;
    }
    const float inv = 1.f / sum;
    #pragma unroll
    for (int k = 0; k < TOPK; ++k) s_attn[f * TOPK + k] = tmp[k] * inv;
  }
  __syncthreads();

  // attention-weighted neighbor aggregation + residual
  float acc = 0.f;
  #pragma unroll
  for (int k = 0; k < TOPK; ++k) {
    const int gnode = b * NNODES + s_idx[k];
    acc += s_attn[head * TOPK + k] * h[(size_t)gnode * FIN + f];
  }
  const float y = acc + x[(size_t)node * FIN + f];

  // LayerNorm over 128 features: wave32 shuffle reduce + LDS combine
  float sum = y;
  for (int off = 16; off > 0; off >>= 1) sum += __shfl_down(sum, off, 32);
  if ((f & 31) == 0) s_red[f >> 5] = sum;
  __syncthreads();
  if (f == 0) s_stat = (s_red[0] + s_red[1] + s_red[2] + s_red[3]) * (1.0f / FIN);
  __syncthreads();
  const float mu = s_stat;
  const float d  = y - mu;

  __syncthreads();                       // protect s_red/s_stat reuse
  float sq = d * d;
  for (int off = 16; off > 0; off >>= 1) sq += __shfl_down(sq, off, 32);
  if ((f & 31) == 0) s_red[f >> 5] = sq;
  __syncthreads();
  if (f == 0) s_stat = (s_red[0] + s_red[1] + s_red[2] + s_red[3]) * (1.0f / FIN);
  __syncthreads();
  const float var = s_stat;

  out[(size_t)node * FIN + f] = d * rsqrtf(var + LN_EPS) * gamma[f] + beta[f];
}

// ---------------------------------------------------------------------------
extern "C" void kernel_launch(void* const* d_in, const int* in_sizes, int n_in,
                              void* d_out, int out_size, void* d_ws, size_t ws_size,
                              hipStream_t stream) {
  const float* x         = (const float*)d_in[0];
  const float* mask      = (const float*)d_in[1];
  const float* positions = (const float*)d_in[2];
  const float* W         = (const float*)d_in[3];
  const float* a_src     = (const float*)d_in[4];
  const float* a_dst     = (const float*)d_in[5];
  const float* ln_gamma  = (const float*)d_in[6];
  const float* ln_beta   = (const float*)d_in[7];
  float* out = (float*)d_out;

  const size_t BN = (size_t)BATCH * NNODES;      // 32768
  float* h   = (float*)d_ws;                     // BN*FIN        (16 MB)
  float* e_i = h   + BN * FIN;                   // BN*NH         (0.5 MB)
  float* e_j = e_i + BN * NH;                    // BN*NH         (0.5 MB)
  int*   idx = (int*)(e_j + BN * NH);            // BN*TOPK       (2 MB)

  gemm_xw_wmma   <<<2048, 256, 0, stream>>>(x, W, h);
  compute_e      <<<(int)((BN * NH) / 256), 256, 0, stream>>>(h, a_src, a_dst, e_i, e_j);
  topk_kernel    <<<BATCH * 16, 256, 0, stream>>>(positions, idx);
  aggregate_kernel<<<(int)BN, 128, 0, stream>>>(x, mask, h, e_i, e_j, idx,
                                                ln_gamma, ln_beta, out);
}